// LinearLoss_31190052503810
// MI455X (gfx1250) — compile-verified
//
#include <hip/hip_runtime.h>

// LinearLoss on MI455X (gfx1250, wave32).
// Bandwidth-bound: 402 MB streamed -> ~17us at 23.3 TB/s. Compute is trivial;
// the 128x256x1024 f32 GEMM is done with V_WMMA_F32_16X16X4_F32.

typedef __attribute__((ext_vector_type(2))) float v2f;
typedef __attribute__((ext_vector_type(8))) float v8f;

#define SEG_ROWS 256
#define SEG_COLS 1024

// ---------------- Kernel 0: zero scratch seg matrix + loss scalar ----------
__global__ void k_zero(float* __restrict__ seg, float* __restrict__ out) {
    int t = blockIdx.x * blockDim.x + threadIdx.x;
    if (t < SEG_ROWS * SEG_COLS) seg[t] = 0.0f;
    if (t == 0) out[0] = 0.0f;
}

// ---------------- Kernel 1: mu0 reduce(axis=2) + segment scatter -----------
// seg[idx0[j]][i] += sum_k mu0[i][j][k]   (k = 0..31 contiguous -> 128B/thread)
__global__ void __launch_bounds__(256)
k_mu0_scatter(const float* __restrict__ mu0, const int* __restrict__ idx0,
              float* __restrict__ seg) {
    int p = blockIdx.x * blockDim.x + threadIdx.x;  // 0 .. 1024*1024-1
    int i = p >> 10;        // first axis of mu0
    int j = p & 1023;       // second axis (scatter key)
    const float4* src = (const float4*)(mu0 + (size_t)p * 32);
    float s = 0.0f;
#pragma unroll
    for (int q = 0; q < 8; ++q) {       // 8 x global_load_b128, fully coalesced
        float4 v = src[q];
        s += (v.x + v.y) + (v.z + v.w);
    }
    atomicAdd(&seg[idx0[j] * SEG_COLS + i], s);  // global_atomic_add_f32
}

// ---------------- Kernel 2: WMMA GEMM + loss0 epilogue ---------------------
// proc0 = mapping(128x256) @ seg(256x1024); loss += 0.5*||obs0 - proc0||^2
// One wave32 per 16x16 output tile; 8 M-tiles x 64 N-tiles = 512 waves.
// f32 A 16x4 layout (ISA 7.12.2): lanes 0-15 hold K={0,1}, lanes 16-31 K={2,3}.
__global__ void __launch_bounds__(256)
k_gemm_loss0(const float* __restrict__ mapping, const float* __restrict__ seg,
             const float* __restrict__ obs0, float* __restrict__ out) {
    __shared__ float red[256];
    const int lane = threadIdx.x & 31;
    const int wave = threadIdx.x >> 5;
    const int gw   = blockIdx.x * 8 + wave;   // 0..511
    const int tm   = gw >> 6;                 // 0..7   (M tile)
    const int tn   = gw & 63;                 // 0..63  (N tile)

    const int half = lane >> 4;               // 0: K pair {0,1}; 1: K pair {2,3}
    const int l15  = lane & 15;
    const int rowA = tm * 16 + l15;           // A-matrix M index for this lane
    const int col  = tn * 16 + l15;           // B/C/D N index for this lane

    v8f c = {0.f, 0.f, 0.f, 0.f, 0.f, 0.f, 0.f, 0.f};
    for (int k0 = 0; k0 < 256; k0 += 4) {
        const int kb = k0 + half * 2;
        // A fragment: 2 consecutive K values of one row of mapping
        v2f a;
        {
            float2 av = *(const float2*)(mapping + rowA * 256 + kb);  // kb even
            a.x = av.x; a.y = av.y;
        }
        // B fragment: rows kb, kb+1 of seg at column `col`
        v2f b;
        b.x = seg[kb * SEG_COLS + col];
        b.y = seg[(kb + 1) * SEG_COLS + col];
        // D = A x B + C   (v_wmma_f32_16x16x4_f32)
        c = __builtin_amdgcn_wmma_f32_16x16x4_f32(
                /*neg_a=*/false, a, /*neg_b=*/false, b,
                /*c_mod=*/(short)0, c, /*reuse_a=*/false, /*reuse_b=*/false);
    }

    // C/D layout: VGPR r -> M = r (lanes 0-15) or 8+r (lanes 16-31), N = lane%16
    const int mBase = tm * 16 + half * 8;
    float part = 0.0f;
#pragma unroll
    for (int r = 0; r < 8; ++r) {
        float d = obs0[(mBase + r) * 1024 + col] - c[r];
        part += 0.5f * d * d;
    }
    red[threadIdx.x] = part;
    __syncthreads();
    for (int s = 128; s > 0; s >>= 1) {
        if (threadIdx.x < s) red[threadIdx.x] += red[threadIdx.x + s];
        __syncthreads();
    }
    if (threadIdx.x == 0) atomicAdd(out, red[0]);
}

// ---------------- Kernel 3: mu1 reduce(axis=1) + loss1 ---------------------
// proc1[b][k] = sum_m mu1[b][m][k]; loss += 0.45*||obs1 - proc1||^2
// One block per b; 16 m-groups x 16 k-quads; float4 loads (256B/row coalesced).
__global__ void __launch_bounds__(256)
k_mu1_loss1(const float* __restrict__ mu1, const float* __restrict__ obs1,
            float* __restrict__ out) {
    __shared__ float4 sh[256];
    __shared__ float red[16];
    const int b  = blockIdx.x;           // 0..2047
    const int tq = threadIdx.x & 15;     // k-quad: covers k = tq*4 .. tq*4+3
    const int mg = threadIdx.x >> 4;     // m-group 0..15
    const float* base = mu1 + (size_t)b * 512 * 64;

    float4 acc = {0.f, 0.f, 0.f, 0.f};
    for (int m = mg; m < 512; m += 16) {
        float4 v = *(const float4*)(base + m * 64 + tq * 4);
        acc.x += v.x; acc.y += v.y; acc.z += v.z; acc.w += v.w;
    }
    sh[threadIdx.x] = acc;
    __syncthreads();
    if (threadIdx.x < 16) {
        const int t = threadIdx.x;
        float4 s = {0.f, 0.f, 0.f, 0.f};
#pragma unroll
        for (int g = 0; g < 16; ++g) {
            float4 v = sh[g * 16 + t];
            s.x += v.x; s.y += v.y; s.z += v.z; s.w += v.w;
        }
        const float* o = obs1 + b * 64 + t * 4;
        float d0 = o[0] - s.x, d1 = o[1] - s.y;
        float d2 = o[2] - s.z, d3 = o[3] - s.w;
        red[t] = 0.45f * (d0 * d0 + d1 * d1 + d2 * d2 + d3 * d3);  // 0.9 * 0.5
    }
    __syncthreads();
    if (threadIdx.x == 0) {
        float tot = 0.0f;
#pragma unroll
        for (int t = 0; t < 16; ++t) tot += red[t];
        atomicAdd(out, tot);
    }
}

// ---------------- Launcher -------------------------------------------------
extern "C" void kernel_launch(void* const* d_in, const int* in_sizes, int n_in,
                              void* d_out, int out_size, void* d_ws, size_t ws_size,
                              hipStream_t stream) {
    const float* mu0     = (const float*)d_in[0];  // (1024,1024,32)
    const float* mu1     = (const float*)d_in[1];  // (2048,512,64)
    const float* obs0    = (const float*)d_in[2];  // (128,1024)
    const float* obs1    = (const float*)d_in[3];  // (2048,64)
    const float* mapping = (const float*)d_in[4];  // (128,256)
    const int*   idx0    = (const int*)d_in[5];    // (1024,)
    float* out = (float*)d_out;
    float* seg = (float*)d_ws;                     // 256*1024 f32 = 1 MB scratch

    k_zero<<<(SEG_ROWS * SEG_COLS + 255) / 256, 256, 0, stream>>>(seg, out);
    k_mu0_scatter<<<(1024 * 1024) / 256, 256, 0, stream>>>(mu0, idx0, seg);
    k_gemm_loss0<<<64, 256, 0, stream>>>(mapping, seg, obs0, out);
    k_mu1_loss1<<<2048, 256, 0, stream>>>(mu1, obs1, out);
}